// HouseholdEncoder_46952582480179
// MI455X (gfx1250) — compile-verified
//
#include <hip/hip_runtime.h>
#include <hip/hip_bf16.h>
#include <math.h>

// ---------------------------------------------------------------------------
// HouseholdEncoder (3-layer GAT + softmax pooling) for MI455X / gfx1250.
// GEMMs use V_WMMA_F32_16X16X4_F32 (exact f32 semantics; GEMM is not the
// bottleneck here — the edge gather/scatter phase is, and it is L2-resident
// because h (51.2MB) < 192MB L2).
// ---------------------------------------------------------------------------

#define N_NODES   50000
#define N_EDGES   300000
#define IN_DIM    25
#define HID       64
#define HEADS     4
#define N_LAYERS  3
#define NEG_SLOPE 0.2f
#define HOUT      (HID * HEADS)   // 256

typedef __attribute__((ext_vector_type(2))) float v2f;
typedef __attribute__((ext_vector_type(8))) float v8f;

// ---------------- helpers ----------------

__device__ __forceinline__ void atomicMaxFloat(float* addr, float val) {
    if (val >= 0.0f) atomicMax((int*)addr, __float_as_int(val));
    else             atomicMin((unsigned int*)addr, __float_as_uint(val));
}

// ---------------- kernels ----------------

// x[n,j] = sum_k nf[n,k]*Wp[k,j] + bp[j]        (tiny GEMM, VALU is fine)
__global__ void k_proj(const float* __restrict__ nf, const float* __restrict__ Wp,
                       const float* __restrict__ bp, float* __restrict__ x) {
    int gid = blockIdx.x * blockDim.x + threadIdx.x;
    if (gid >= N_NODES * HID) return;
    int n = gid >> 6, j = gid & 63;
    float acc = bp[j];
    const float* row = nf + n * IN_DIM;
#pragma unroll
    for (int k = 0; k < IN_DIM; ++k) acc = fmaf(row[k], Wp[k * HID + j], acc);
    x[gid] = acc;
}

// H[M,256] = X[M,64] @ W[64,256] via V_WMMA_F32_16X16X4_F32.
// One wave computes one 16x16 tile; K=64 -> 16 chained WMMAs.
// A 16x4 f32 layout: lanes 0-15 hold K=0,1 (v0,v1); lanes 16-31 hold K=2,3.
// C/D: lane<16 -> N=lane, M=vgpr; lane>=16 -> N=lane-16, M=vgpr+8.
__global__ void __launch_bounds__(256)
k_gemm_wmma(const float* __restrict__ X, const float* __restrict__ W,
            float* __restrict__ H) {
    const int wave   = (blockIdx.x * blockDim.x + threadIdx.x) >> 5;
    const int lane   = threadIdx.x & 31;
    const int n_tile = wave & 15;          // 256/16 = 16 N-tiles
    const int m_tile = wave >> 4;          // 3125 M-tiles (50000/16 exact)
    const int m0 = m_tile * 16, n0 = n_tile * 16;
    const int half = lane >> 4;            // 0 or 1
    const int l    = lane & 15;

    const float* Arow = X + (m0 + l) * HID;      // this lane's A row
    const float* Bcol = W + n0 + l;              // this lane's B column

    v8f c = {};
#pragma unroll
    for (int k = 0; k < HID; k += 4) {
        const int ka = k + 2 * half;
        v2f a = { Arow[ka], Arow[ka + 1] };
        v2f b = { Bcol[(size_t)ka * HOUT], Bcol[(size_t)(ka + 1) * HOUT] };
        c = __builtin_amdgcn_wmma_f32_16x16x4_f32(
                /*neg_a=*/false, a, /*neg_b=*/false, b,
                /*c_mod=*/(short)0, c, /*reuse_a=*/false, /*reuse_b=*/false);
    }
#pragma unroll
    for (int v = 0; v < 8; ++v)
        H[(size_t)(m0 + v + 8 * half) * HOUT + (n0 + l)] = c[v];
}

// Per-node attention projections: s_src[n,h] = <h[n,h,:], a_src[h]>, same for tgt.
__global__ void k_scores(const float* __restrict__ H, const float* __restrict__ a,
                         float* __restrict__ s_src, float* __restrict__ s_tgt) {
    int gid = blockIdx.x * blockDim.x + threadIdx.x;
    if (gid >= N_NODES * HEADS) return;
    int n = gid >> 2, hd = gid & 3;
    const float* hrow = H + (size_t)n * HOUT + hd * HID;
    const float* as   = a + hd * (2 * HID);
    const float* at   = as + HID;
    float vs = 0.f, vt = 0.f;
#pragma unroll 8
    for (int f = 0; f < HID; ++f) {
        float hv = hrow[f];
        vs = fmaf(hv, as[f], vs);
        vt = fmaf(hv, at[f], vt);
    }
    s_src[gid] = vs;
    s_tgt[gid] = vt;
}

// Zero the scatter target and reset per-head softmax scalars.
__global__ void k_init_layer(float* __restrict__ xout, float* __restrict__ scal) {
    int gid = blockIdx.x * blockDim.x + threadIdx.x;
    for (int i = gid; i < N_NODES * HID; i += gridDim.x * blockDim.x) xout[i] = 0.f;
    if (gid < HEADS)                 scal[gid] = -INFINITY;   // head_max
    else if (gid < 2 * HEADS)        scal[gid] = 0.f;         // head_sum
}

// alpha[e,h] = leaky_relu(s_src[src]+s_tgt[tgt]); track per-head global max.
__global__ void k_alpha(const float* __restrict__ s_src, const float* __restrict__ s_tgt,
                        const long long* __restrict__ src, const long long* __restrict__ tgt,
                        float* __restrict__ alpha, float* __restrict__ head_max) {
    __shared__ float red[256];
    float lmax[HEADS] = { -INFINITY, -INFINITY, -INFINITY, -INFINITY };
    const int stride = gridDim.x * blockDim.x;
    for (int e = blockIdx.x * blockDim.x + threadIdx.x; e < N_EDGES; e += stride) {
        int s = (int)src[e], t = (int)tgt[e];
#pragma unroll
        for (int hd = 0; hd < HEADS; ++hd) {
            float v = s_src[s * HEADS + hd] + s_tgt[t * HEADS + hd];
            v = v > 0.f ? v : NEG_SLOPE * v;
            alpha[(size_t)e * HEADS + hd] = v;
            lmax[hd] = fmaxf(lmax[hd], v);
        }
    }
#pragma unroll
    for (int hd = 0; hd < HEADS; ++hd) {
        red[threadIdx.x] = lmax[hd];
        __syncthreads();
        for (int off = 128; off > 0; off >>= 1) {
            if ((int)threadIdx.x < off)
                red[threadIdx.x] = fmaxf(red[threadIdx.x], red[threadIdx.x + off]);
            __syncthreads();
        }
        if (threadIdx.x == 0) atomicMaxFloat(&head_max[hd], red[0]);
        __syncthreads();
    }
}

// alpha <- exp(alpha - head_max); accumulate per-head global sum.
__global__ void k_expsum(float* __restrict__ alpha, const float* __restrict__ head_max,
                         float* __restrict__ head_sum) {
    __shared__ float red[256];
    float lsum[HEADS] = { 0.f, 0.f, 0.f, 0.f };
    float hmax[HEADS];
#pragma unroll
    for (int hd = 0; hd < HEADS; ++hd) hmax[hd] = head_max[hd];
    const int stride = gridDim.x * blockDim.x;
    for (int e = blockIdx.x * blockDim.x + threadIdx.x; e < N_EDGES; e += stride) {
#pragma unroll
        for (int hd = 0; hd < HEADS; ++hd) {
            float w = __expf(alpha[(size_t)e * HEADS + hd] - hmax[hd]);
            alpha[(size_t)e * HEADS + hd] = w;
            lsum[hd] += w;
        }
    }
#pragma unroll
    for (int hd = 0; hd < HEADS; ++hd) {
        red[threadIdx.x] = lsum[hd];
        __syncthreads();
        for (int off = 128; off > 0; off >>= 1) {
            if ((int)threadIdx.x < off) red[threadIdx.x] += red[threadIdx.x + off];
            __syncthreads();
        }
        if (threadIdx.x == 0) atomicAdd(&head_sum[hd], red[0]);
        __syncthreads();
    }
}

// Message pass: xout[tgt,j] += sum_h (alpha/head_sum) * 0.25 * h[src,h,j]
// (head-mean folded into the scatter). One thread per (edge, j).
__global__ void k_scatter(const float* __restrict__ alpha, const float* __restrict__ head_sum,
                          const float* __restrict__ H,
                          const long long* __restrict__ src, const long long* __restrict__ tgt,
                          float* __restrict__ xout) {
    int gid = blockIdx.x * blockDim.x + threadIdx.x;
    if (gid >= N_EDGES * HID) return;
    int e = gid >> 6, j = gid & 63;
    int s = (int)src[e], t = (int)tgt[e];
    const float* hrow = H + (size_t)s * HOUT + j;
    float acc = 0.f;
#pragma unroll
    for (int hd = 0; hd < HEADS; ++hd) {
        float coef = alpha[(size_t)e * HEADS + hd] / head_sum[hd] * (1.0f / HEADS);
        acc = fmaf(coef, hrow[hd * HID], acc);
    }
    atomicAdd(&xout[(size_t)t * HID + j], acc);
}

__global__ void k_elu(float* __restrict__ x) {
    int gid = blockIdx.x * blockDim.x + threadIdx.x;
    if (gid >= N_NODES * HID) return;
    float v = x[gid];
    x[gid] = v > 0.f ? v : expm1f(v);
}

// ---------------- pooling ----------------

__global__ void k_pool_init(float* __restrict__ scal, float* __restrict__ out) {
    if (threadIdx.x < HID) out[threadIdx.x] = 0.f;
    if (threadIdx.x == 0) { scal[8] = -INFINITY; scal[9] = 0.f; }
}

__global__ void k_pool_score(const float* __restrict__ x, const float* __restrict__ Wpool,
                             const float* __restrict__ bpool,
                             float* __restrict__ scores, float* __restrict__ scal) {
    __shared__ float red[256];
    float lmax = -INFINITY;
    const int stride = gridDim.x * blockDim.x;
    for (int n = blockIdx.x * blockDim.x + threadIdx.x; n < N_NODES; n += stride) {
        const float* row = x + (size_t)n * HID;
        float acc = bpool[0];
#pragma unroll 8
        for (int f = 0; f < HID; ++f) acc = fmaf(row[f], Wpool[f], acc);
        scores[n] = acc;
        lmax = fmaxf(lmax, acc);
    }
    red[threadIdx.x] = lmax;
    __syncthreads();
    for (int off = 128; off > 0; off >>= 1) {
        if ((int)threadIdx.x < off)
            red[threadIdx.x] = fmaxf(red[threadIdx.x], red[threadIdx.x + off]);
        __syncthreads();
    }
    if (threadIdx.x == 0) atomicMaxFloat(&scal[8], red[0]);
}

__global__ void k_pool_exp(float* __restrict__ scores, float* __restrict__ scal) {
    __shared__ float red[256];
    const float smax = scal[8];
    float lsum = 0.f;
    const int stride = gridDim.x * blockDim.x;
    for (int n = blockIdx.x * blockDim.x + threadIdx.x; n < N_NODES; n += stride) {
        float w = __expf(scores[n] - smax);
        scores[n] = w;
        lsum += w;
    }
    red[threadIdx.x] = lsum;
    __syncthreads();
    for (int off = 128; off > 0; off >>= 1) {
        if ((int)threadIdx.x < off) red[threadIdx.x] += red[threadIdx.x + off];
        __syncthreads();
    }
    if (threadIdx.x == 0) atomicAdd(&scal[9], red[0]);
}

// emb[j] = sum_n (scores[n]/ssum) * x[n,j]  (64 threads/block, block-partial sums)
__global__ void k_pool_emb(const float* __restrict__ x, const float* __restrict__ scores,
                           const float* __restrict__ scal, float* __restrict__ out) {
    const int j = threadIdx.x;                // 0..63
    const float inv = 1.0f / scal[9];
    float acc = 0.f;
    for (int n = blockIdx.x; n < N_NODES; n += gridDim.x)
        acc = fmaf(scores[n], x[(size_t)n * HID + j], acc);
    atomicAdd(&out[j], acc * inv);
}

// ---------------- launch ----------------

extern "C" void kernel_launch(void* const* d_in, const int* in_sizes, int n_in,
                              void* d_out, int out_size, void* d_ws, size_t ws_size,
                              hipStream_t stream) {
    const float*     nf      = (const float*)d_in[0];
    const long long* ei      = (const long long*)d_in[1];   // int64 [2, N_EDGES]
    const float*     Wp      = (const float*)d_in[2];
    const float*     bp      = (const float*)d_in[3];
    const float*     layer_W = (const float*)d_in[4];       // [3,64,256]
    const float*     layer_a = (const float*)d_in[5];       // [3,4,128]
    const float*     Wpool   = (const float*)d_in[6];
    const float*     bpool   = (const float*)d_in[7];
    float*           out     = (float*)d_out;               // [64 + 50000*64]

    const long long* src = ei;
    const long long* tgt = ei + N_EDGES;

    // workspace layout (bytes)
    char* ws = (char*)d_ws;
    const size_t SZ_X = (size_t)N_NODES * HID * sizeof(float);    // 12.8 MB
    float* X0    = (float*)(ws);
    float* X1    = (float*)(ws + SZ_X);
    float* H     = (float*)(ws + 2 * SZ_X);                       // 51.2 MB
    float* Ssrc  = (float*)(ws + 2 * SZ_X + (size_t)N_NODES * HOUT * sizeof(float));
    float* Stgt  = Ssrc + (size_t)N_NODES * HEADS;
    float* Alpha = Stgt + (size_t)N_NODES * HEADS;                // 4.8 MB
    float* Scal  = Alpha + (size_t)N_EDGES * HEADS;               // [0..3] hmax [4..7] hsum [8] smax [9] ssum
    float* Scores= Scal + 16;                                     // [N_NODES]
    float* head_max = Scal;
    float* head_sum = Scal + HEADS;

    const int T = 256;
    const int grid_nodes = (N_NODES * HID + T - 1) / T;           // 12500
    const int grid_gemm  = (N_NODES / 16) * (HOUT / 16) / (T / 32); // 3125*16/8 = 6250
    const int grid_sc    = (N_NODES * HEADS + T - 1) / T;
    const int grid_scat  = (N_EDGES * HID + T - 1) / T;           // 75000

    // input projection
    k_proj<<<grid_nodes, T, 0, stream>>>(nf, Wp, bp, X0);

    float* xin = X0;
    float* xout = X1;
    for (int i = 0; i < N_LAYERS; ++i) {
        k_gemm_wmma<<<grid_gemm, T, 0, stream>>>(xin, layer_W + (size_t)i * HID * HOUT, H);
        k_scores   <<<grid_sc,   T, 0, stream>>>(H, layer_a + (size_t)i * HEADS * 2 * HID, Ssrc, Stgt);
        k_init_layer<<<grid_nodes, T, 0, stream>>>(xout, Scal);
        k_alpha    <<<512, T, 0, stream>>>(Ssrc, Stgt, src, tgt, Alpha, head_max);
        k_expsum   <<<512, T, 0, stream>>>(Alpha, head_max, head_sum);
        k_scatter  <<<grid_scat, T, 0, stream>>>(Alpha, head_sum, H, src, tgt, xout);
        k_elu      <<<grid_nodes, T, 0, stream>>>(xout);
        float* tmp = xin; xin = xout; xout = tmp;
    }

    // softmax pooling over nodes; outputs: emb[0:64], x[64:]
    k_pool_init <<<1, 64, 0, stream>>>(Scal, out);
    k_pool_score<<<256, T, 0, stream>>>(xin, Wpool, bpool, Scores, Scal);
    k_pool_exp  <<<256, T, 0, stream>>>(Scores, Scal);
    k_pool_emb  <<<256, 64, 0, stream>>>(xin, Scores, Scal, out);
    hipMemcpyAsync(out + HID, xin, SZ_X, hipMemcpyDeviceToDevice, stream);
}